// StationFlowGNN_24532853195354
// MI455X (gfx1250) — compile-verified
//
#include <hip/hip_runtime.h>
#include <hip/hip_bf16.h>

typedef float v2f __attribute__((ext_vector_type(2)));
typedef float v8f __attribute__((ext_vector_type(8)));

// ---------------------------------------------------------------------------
// Small elementwise kernels
// ---------------------------------------------------------------------------

__global__ void gcn_fill(float* __restrict__ p, float v, int n) {
    for (int i = blockIdx.x * blockDim.x + threadIdx.x; i < n;
         i += gridDim.x * blockDim.x)
        p[i] = v;
}

// deg[dst[e]] += 1 for every real edge (deg pre-initialized to 1.0 = self loop)
__global__ void gcn_deg(const long long* __restrict__ dst,
                        float* __restrict__ deg, int E) {
    for (int e = blockIdx.x * blockDim.x + threadIdx.x; e < E;
         e += gridDim.x * blockDim.x)
        atomicAdd(&deg[(int)dst[e]], 1.0f);
}

__global__ void gcn_rsqrt(const float* __restrict__ deg,
                          float* __restrict__ dis, int n) {
    for (int i = blockIdx.x * blockDim.x + threadIdx.x; i < n;
         i += gridDim.x * blockDim.x) {
        float d = deg[i];
        dis[i] = d > 0.0f ? rsqrtf(d) : 0.0f;
    }
}

// agg[i,f] = dis[i]^2 * h[i,f] + b[f]   (self-loop contribution + bias)
__global__ void gcn_agg_init(const float* __restrict__ h,
                             const float* __restrict__ dis,
                             const float* __restrict__ b,
                             float* __restrict__ agg, int n) {
    const int total = n * 128;
    for (int idx = blockIdx.x * blockDim.x + threadIdx.x; idx < total;
         idx += gridDim.x * blockDim.x) {
        int i = idx >> 7;
        int f = idx & 127;
        float di = dis[i];
        agg[idx] = di * di * h[idx] + b[f];
    }
}

// One wave (32 lanes) per edge; lane l handles features [4l, 4l+4).
// agg[dst,:] += dis[src]*dis[dst] * h[src,:]
__global__ void gcn_edge_agg(const long long* __restrict__ src,
                             const long long* __restrict__ dst,
                             const float* __restrict__ dis,
                             const float* __restrict__ h,
                             float* __restrict__ agg, int E) {
    const int lane   = threadIdx.x & 31;
    const int wid    = (blockIdx.x * blockDim.x + threadIdx.x) >> 5;
    const int nwaves = (gridDim.x * blockDim.x) >> 5;
    for (int e = wid; e < E; e += nwaves) {
        const int s = (int)src[e];
        const int d = (int)dst[e];
        const float c = dis[s] * dis[d];
        const float4 v =
            *reinterpret_cast<const float4*>(h + (size_t)s * 128 + lane * 4);
        float* o = agg + (size_t)d * 128 + lane * 4;
        atomicAdd(o + 0, c * v.x);
        atomicAdd(o + 1, c * v.y);
        atomicAdd(o + 2, c * v.z);
        atomicAdd(o + 3, c * v.w);
    }
}

// ---------------------------------------------------------------------------
// WMMA fp32 GEMM:  C[M x N] = (RELU? relu(A) : A)[M x 128] * W[128 x N] (+bias)
// NT waves per block; wave w owns output columns [16w, 16w+16).
// Each wave keeps its whole B panel (128 x 16 of W) in 64 VGPRs.
// A tile (16 x 128) staged in LDS with padded stride for conflict-free reads.
//
// V_WMMA_F32_16X16X4_F32 VGPR layout (ISA 7.12.2):
//   A (16x4): lane r (r<16) v0=A[r][0],v1=A[r][1]; lanes 16-31: K=2,3
//   B (4x16): v0: K=0 (lanes 0-15, N=lane) / K=2 (lanes 16-31)
//             v1: K=1 / K=3
//   C/D     : vgpr i: M=i (lanes 0-15) / M=i+8 (lanes 16-31), N=lane%16
// ---------------------------------------------------------------------------
template <int NT, bool RELU, bool BIAS>
__global__ __launch_bounds__(NT * 32) void gcn_gemm(
    const float* __restrict__ A, const float* __restrict__ W,
    const float* __restrict__ bias, float* __restrict__ C, int M) {
    constexpr int N   = NT * 16;  // output width
    constexpr int K   = 128;      // reduction dim (F == H == 128)
    constexpr int LDA = K + 4;    // padded LDS row stride (bank-conflict free)
    __shared__ float sA[16 * LDA];

    const int tid  = threadIdx.x;
    const int wave = tid >> 5;
    const int lane = tid & 31;
    const int half = lane >> 4;  // 0: lanes 0-15, 1: lanes 16-31
    const int r    = lane & 15;
    const int n0   = wave * 16;

    // Preload this wave's B fragments: W[:, n0:n0+16] -> 32 x v2f (64 VGPRs).
    v2f bfrag[K / 4];
#pragma unroll
    for (int k = 0; k < K / 4; ++k) {
        const int row = 4 * k + 2 * half;
        bfrag[k][0] = W[(size_t)row * N + n0 + r];
        bfrag[k][1] = W[(size_t)(row + 1) * N + n0 + r];
    }

    for (int mt = blockIdx.x; mt * 16 < M; mt += gridDim.x) {
        const int m0 = mt * 16;

        __syncthreads();
        // Stage A tile (coalesced), fuse ReLU, zero-pad tail rows.
        for (int idx = tid; idx < 16 * K; idx += NT * 32) {
            const int row = idx >> 7;
            const int col = idx & 127;
            float v = 0.0f;
            if (m0 + row < M) {
                v = A[(size_t)(m0 + row) * K + col];
                if (RELU) v = v > 0.0f ? v : 0.0f;
            }
            sA[row * LDA + col] = v;
        }
        __syncthreads();

        v8f acc = {};
#pragma unroll
        for (int k = 0; k < K / 4; ++k) {
            v2f afrag;
            const float* ap = &sA[r * LDA + 4 * k + 2 * half];
            afrag[0] = ap[0];
            afrag[1] = ap[1];
            acc = __builtin_amdgcn_wmma_f32_16x16x4_f32(
                false, afrag, false, bfrag[k], (short)0, acc, false, false);
        }

#pragma unroll
        for (int i = 0; i < 8; ++i) {
            const int row = m0 + i + 8 * half;
            if (row < M) {
                float v = acc[i];
                if (BIAS) v += bias[n0 + r];
                C[(size_t)row * N + n0 + r] = v;
            }
        }
    }
}

// ---------------------------------------------------------------------------
// Driver
// ---------------------------------------------------------------------------
extern "C" void kernel_launch(void* const* d_in, const int* in_sizes, int n_in,
                              void* d_out, int out_size, void* d_ws,
                              size_t ws_size, hipStream_t stream) {
    const float*     x   = (const float*)d_in[0];
    const long long* ei  = (const long long*)d_in[1];  // int64 [2, E]
    const float*     W1  = (const float*)d_in[2];
    const float*     b1  = (const float*)d_in[3];
    const float*     W2  = (const float*)d_in[4];
    const float*     b2  = (const float*)d_in[5];
    const float*     Wfc = (const float*)d_in[6];
    const float*     bfc = (const float*)d_in[7];
    float*           out = (float*)d_out;

    const int H  = in_sizes[3];       // 128
    const int Nn = in_sizes[0] / H;   // 50000 nodes (F == H)
    const int E  = in_sizes[1] / 2;   // 800000 edges
    (void)n_in; (void)out_size; (void)ws_size;

    // Workspace layout: deg[Nn] | dis[Nn] | bufA[Nn*H] | bufB[Nn*H]
    float* deg  = (float*)d_ws;
    float* dis  = deg + Nn;
    float* bufA = dis + Nn;
    float* bufB = bufA + (size_t)Nn * H;

    const long long* srcp = ei;
    const long long* dstp = ei + E;

    const int T        = 256;
    const int nbNodes  = (Nn + T - 1) / T;
    const int nbEdges  = (E + T - 1) / T;
    const int nbElems  = (int)(((size_t)Nn * H + T - 1) / T);
    const int nbEdgeW  = 6144;  // grid-stride, 8 waves/block, 1 edge/wave
    const int mtiles   = (Nn + 15) / 16;
    const int nbGemm8  = mtiles < 1536 ? mtiles : 1536;
    const int nbGemm4  = mtiles < 2048 ? mtiles : 2048;

    // Degrees (self-loop => init 1.0), then dis = deg^{-1/2}
    gcn_fill<<<nbNodes, T, 0, stream>>>(deg, 1.0f, Nn);
    gcn_deg<<<nbEdges, T, 0, stream>>>(dstp, deg, E);
    gcn_rsqrt<<<nbNodes, T, 0, stream>>>(deg, dis, Nn);

    // Layer 1: h1 = x @ W1 ; agg1 = selfloop + b1 + edge scatter
    gcn_gemm<8, false, false><<<nbGemm8, 256, 0, stream>>>(x, W1, nullptr, bufA, Nn);
    gcn_agg_init<<<nbElems, T, 0, stream>>>(bufA, dis, b1, bufB, Nn);
    gcn_edge_agg<<<nbEdgeW, T, 0, stream>>>(srcp, dstp, dis, bufA, bufB, E);

    // Layer 2: h2 = relu(agg1) @ W2 ; agg2 = selfloop + b2 + edge scatter
    gcn_gemm<8, true, false><<<nbGemm8, 256, 0, stream>>>(bufB, W2, nullptr, bufA, Nn);
    gcn_agg_init<<<nbElems, T, 0, stream>>>(bufA, dis, b2, bufB, Nn);
    gcn_edge_agg<<<nbEdgeW, T, 0, stream>>>(srcp, dstp, dis, bufA, bufB, E);

    // Head: out = relu(agg2) @ Wfc + bfc   (N = 64 -> 4 waves/block)
    gcn_gemm<4, true, true><<<nbGemm4, 128, 0, stream>>>(bufB, Wfc, bfc, out, Nn);
}